// RidgeRegressor_9242769622530
// MI455X (gfx1250) — compile-verified
//
#include <hip/hip_runtime.h>
#include <hip/hip_bf16.h>

typedef __attribute__((ext_vector_type(2))) float v2f;
typedef __attribute__((ext_vector_type(8))) float v8f;

#define NBATCH 64
#define NSAMP  2048
#define NDIM   256
#define ODIM   64
#define ZC     320          // 256 reprs cols + 64 x cols
#define LSTR   336          // LDS row stride (dwords): 2*336 mod 64 == 32 -> no half-wave bank overlap
#define LDA    260          // padded leading dim of A (257x257)
#define NTRI   33153        // 257*258/2 packed lower triangle
#define NYEL   16448        // 257*64

// ---------------------------------------------------------------------------
// Kernel 1: P = X^T [X | x] per batch via V_WMMA_F32_16X16X4_F32, plus the
// "ones" row/column sums (folded into row-block 0 from the staged LDS tile).
// grid = (8 row-blocks of 32 dims, 64 batches), block = 256 threads = 8 waves.
// ---------------------------------------------------------------------------
__global__ __launch_bounds__(256) void gram_kernel(const float* __restrict__ reprs,
                                                   const float* __restrict__ xin,
                                                   float* __restrict__ Ab,
                                                   float* __restrict__ Bb) {
  __shared__ float Zs[32 * LSTR];   // 43,008 B

  const int tid  = threadIdx.x;
  const int lane = tid & 31;
  const int wv   = tid >> 5;                 // wave 0..7
  const int b    = blockIdx.y;
  const int ib0  = blockIdx.x * 32;          // dim-row block base (2 I-tiles)

  const float* rB = reprs + (size_t)b * NSAMP * NDIM;
  const float* xB = xin   + (size_t)b * NSAMP * ODIM;

  v8f acc[5] = {};                           // 5 tiles per wave, 8 VGPRs each
  float s0 = 0.0f, s1 = 0.0f;                // column sums (row-block 0 only)

  const int colSel = lane & 15;              // M/N column within tile
  const int rowSel = (lane >> 4) * 2;        // K offset: half-wave split

  for (int kb = 0; kb < NSAMP / 32; ++kb) {
    const int n0 = kb * 32;
    __syncthreads();
    // cooperative stage: 32 rows x 320 cols = 2560 float4
#pragma unroll
    for (int r = 0; r < 10; ++r) {
      int idx = tid + r * 256;
      int n   = idx / 80;
      int c4  = idx - n * 80;
      float4 v;
      if (c4 < 64) v = *(const float4*)(rB + (n0 + n) * NDIM + c4 * 4);
      else         v = *(const float4*)(xB + (n0 + n) * ODIM + (c4 - 64) * 4);
      *(float4*)(&Zs[n * LSTR + c4 * 4]) = v;
    }
    __syncthreads();

    if (blockIdx.x == 0) {                   // fold in column sums for free
#pragma unroll 8
      for (int r = 0; r < 32; ++r) s0 += Zs[r * LSTR + tid];
      if (tid < 64) {
#pragma unroll 8
        for (int r = 0; r < 32; ++r) s1 += Zs[r * LSTR + 256 + tid];
      }
    }

#pragma unroll
    for (int t = 0; t < 5; ++t) {
      const int tl = wv + t * 8;             // tile 0..39
      const int it = tl & 1;                 // which 16-row group
      const int jt = tl >> 1;                // which 16-col group (of 20)
      const int ca = ib0 + it * 16 + colSel; // A gather column (dim)
      const int cb = jt * 16 + colSel;       // B gather column (Z col)
#pragma unroll
      for (int ks = 0; ks < 8; ++ks) {       // 32 samples / K=4
        const int r0 = ks * 4 + rowSel;
        v2f a, bb;
        a.x  = Zs[r0 * LSTR + ca];
        a.y  = Zs[(r0 + 1) * LSTR + ca];
        bb.x = Zs[r0 * LSTR + cb];
        bb.y = Zs[(r0 + 1) * LSTR + cb];
        acc[t] = __builtin_amdgcn_wmma_f32_16x16x4_f32(
            false, a, false, bb, (short)0, acc[t], false, false);
      }
    }
  }

  float* AbB = Ab + (size_t)b * 257 * LDA;
  float* BbB = Bb + (size_t)b * 257 * ODIM;
#pragma unroll
  for (int t = 0; t < 5; ++t) {
    const int tl = wv + t * 8;
    const int it = tl & 1, jt = tl >> 1;
    const int col   = jt * 16 + colSel;
    const int rbase = ib0 + it * 16 + (lane >> 4) * 8;   // D: M = v + 8*(lane/16)
#pragma unroll
    for (int v = 0; v < 8; ++v) {
      const int row = rbase + v;
      const float val = acc[t][v];
      if (col < 256) AbB[row * LDA + col] = val;
      else           BbB[row * ODIM + (col - 256)] = val;
    }
  }
  if (blockIdx.x == 0) {                     // ones row/col of A, last row of B
    AbB[tid * LDA + 256] = s0;
    AbB[256 * LDA + tid] = s0;
    if (tid < 64) BbB[256 * ODIM + tid] = s1;
    if (tid == 0) AbB[256 * LDA + 256] = (float)NSAMP;
  }
}

// ---------------------------------------------------------------------------
// Kernel 2: per-batch Cholesky solve, fully LDS-resident (needs CDNA5's
// 320 KB/WGP LDS: 132.6 KB packed L + 65.8 KB RHS = 198.4 KB, dynamic).
// grid = 64 batches, block = 256 threads.
// ---------------------------------------------------------------------------
__global__ __launch_bounds__(256) void solve_kernel(const float* __restrict__ Ab,
                                                    const float* __restrict__ Bb,
                                                    const float* __restrict__ lamp,
                                                    float* __restrict__ out) {
  extern __shared__ float smem[];
  float* Lp = smem;           // packed lower triangle, NTRI floats
  float* Y  = smem + NTRI;    // 257 x 64 RHS, NYEL floats
  __shared__ float sinv;

  const int tid = threadIdx.x;
  const int b   = blockIdx.x;

  const float lam = lamp[0];
  const float reg = (lam > 20.0f) ? lam : log1pf(expf(lam));   // softplus

  const float* AbB = Ab + (size_t)b * 257 * LDA;
  const float* BbB = Bb + (size_t)b * 257 * ODIM;

  // load packed lower triangle (+ reg on the diagonal)
  for (int idx = tid; idx < NTRI; idx += 256) {
    int i = (int)floorf((sqrtf(8.0f * (float)idx + 1.0f) - 1.0f) * 0.5f);
    while ((i + 1) * (i + 2) / 2 <= idx) ++i;
    while (i * (i + 1) / 2 > idx) --i;
    const int j = idx - i * (i + 1) / 2;
    float v = AbB[i * LDA + j];
    if (i == j) v += reg;
    Lp[idx] = v;
  }
  for (int idx = tid; idx < NYEL; idx += 256) Y[idx] = BbB[idx];
  __syncthreads();

  // right-looking Cholesky (one trailing row per thread per step)
  for (int k = 0; k < 257; ++k) {
    const int kk = k * (k + 1) / 2 + k;
    if (tid == 0) { const float d = sqrtf(Lp[kk]); Lp[kk] = d; sinv = 1.0f / d; }
    __syncthreads();
    const int i = k + 1 + tid;
    if (i < 257) Lp[i * (i + 1) / 2 + k] *= sinv;
    __syncthreads();
    if (i < 257) {
      const int ib = i * (i + 1) / 2;
      const float lik = Lp[ib + k];
      int jb = (k + 1) * (k + 2) / 2;
      for (int j = k + 1; j <= i; ++j) { Lp[ib + j] -= lik * Lp[jb + k]; jb += j + 1; }
    }
    __syncthreads();
  }

  const int o = tid & 63, g = tid >> 6;      // 64 RHS cols x 4 row groups

  // forward: L y = B
  for (int j = 0; j < 257; ++j) {
    const int jj = j * (j + 1) / 2;
    if (tid < 64) Y[j * ODIM + tid] /= Lp[jj + j];
    __syncthreads();
    const float yj = Y[j * ODIM + o];
    for (int i = j + 1 + g; i < 257; i += 4)
      Y[i * ODIM + o] -= Lp[i * (i + 1) / 2 + j] * yj;
    __syncthreads();
  }

  // backward: L^T w = y ; write results as rows finalize
  for (int j = 256; j >= 0; --j) {
    const int jj = j * (j + 1) / 2;
    if (tid < 64) {
      const float v = Y[j * ODIM + tid] / Lp[jj + j];
      Y[j * ODIM + tid] = v;
      if (j < 256) out[(size_t)b * NDIM * ODIM + j * ODIM + tid] = v;          // w
      else         out[(size_t)NBATCH * NDIM * ODIM + b * ODIM + tid] = v;     // bias
    }
    __syncthreads();
    const float wj = Y[j * ODIM + o];
    for (int i = g; i < j; i += 4)
      Y[i * ODIM + o] -= Lp[jj + i] * wj;    // L^T[i][j] = L[j][i], row-contiguous
    __syncthreads();
  }
}

extern "C" void kernel_launch(void* const* d_in, const int* in_sizes, int n_in,
                              void* d_out, int out_size, void* d_ws, size_t ws_size,
                              hipStream_t stream) {
  const float* reprs = (const float*)d_in[0];
  const float* x     = (const float*)d_in[1];
  const float* lam   = (const float*)d_in[2];
  float* out = (float*)d_out;

  float* Ab = (float*)d_ws;                          // 64 * 257 * 260 f32
  float* Bb = Ab + (size_t)NBATCH * 257 * LDA;       // 64 * 257 * 64  f32

  gram_kernel<<<dim3(8, NBATCH), 256, 0, stream>>>(reprs, x, Ab, Bb);

  const size_t slds = (size_t)(NTRI + NYEL) * sizeof(float);   // 198,404 B (CDNA5 320KB LDS)
  solve_kernel<<<NBATCH, 256, slds, stream>>>(Ab, Bb, lam, out);
}